// MemristorPhysicsLoss_63127429317285
// MI455X (gfx1250) — compile-verified
//
#include <hip/hip_runtime.h>
#include <math.h>

#define BATCHES 64
#define NPG 1024
#define NF 8
#define NTOT (BATCHES * NPG)
#define SLICES 16   // collapse blocks per batch; 128 wave-slots per batch

typedef __attribute__((ext_vector_type(2))) float v2f;
typedef __attribute__((ext_vector_type(8))) float v8f;

// ws layout: [0]=recon_sum [1]=vol_sum [2]=elec_num [3]=elec_cnt [4]=coll_sum [5]=field_sum
__global__ void init_kernel(float* ws) {
    int i = threadIdx.x;
    if (i < 8) ws[i] = 0.0f;
}

// Block-wide reduction (256 threads = 8 waves of 32). op: 0=sum, 1=min, 2=max.
__device__ __forceinline__ float blk_red(float v, int op, float* red8) {
    #pragma unroll
    for (int off = 16; off > 0; off >>= 1) {
        float o = __shfl_down(v, off, 32);
        v = (op == 0) ? (v + o) : ((op == 1) ? fminf(v, o) : fmaxf(v, o));
    }
    int tid = threadIdx.x;
    __syncthreads();
    if ((tid & 31) == 0) red8[tid >> 5] = v;
    __syncthreads();
    float r = red8[0];
    #pragma unroll
    for (int i = 1; i < 8; ++i) {
        float o = red8[i];
        r = (op == 0) ? (r + o) : ((op == 1) ? fminf(r, o) : fmaxf(r, o));
    }
    return r;
}

// One block per batch: recon, volume, electrode, field terms.
__global__ void stats_kernel(const float* __restrict__ pred,
                             const float* __restrict__ tru,
                             const float* __restrict__ orig,
                             const float* __restrict__ nf,
                             float* __restrict__ ws) {
    __shared__ float red8[8];
    const int b   = blockIdx.x;
    const int tid = threadIdx.x;
    const float* pb = pred + (size_t)b * NPG * 3;
    const float* tb = tru  + (size_t)b * NPG * 3;
    const float* ob = orig + (size_t)b * NPG * 3;

    const float BIG = 3.0e38f;
    float pmn[3] = { BIG,  BIG,  BIG};
    float pmx[3] = {-BIG, -BIG, -BIG};
    float omn[3] = { BIG,  BIG,  BIG};
    float omx[3] = {-BIG, -BIG, -BIG};
    float psz = 0.0f, rec = 0.0f;

    for (int i = tid; i < NPG; i += 256) {
        float P[3], O[3], T[3];
        #pragma unroll
        for (int d = 0; d < 3; ++d) {
            P[d] = pb[3 * i + d];
            O[d] = ob[3 * i + d];
            T[d] = tb[3 * i + d];
            pmn[d] = fminf(pmn[d], P[d]);  pmx[d] = fmaxf(pmx[d], P[d]);
            omn[d] = fminf(omn[d], O[d]);  omx[d] = fmaxf(omx[d], O[d]);
            float dr = P[d] - T[d];
            rec += dr * dr;
        }
        psz += P[2];
    }

    float pmnr[3], pmxr[3], omnr[3], omxr[3];
    #pragma unroll
    for (int d = 0; d < 3; ++d) {
        pmnr[d] = blk_red(pmn[d], 1, red8);
        pmxr[d] = blk_red(pmx[d], 2, red8);
        omnr[d] = blk_red(omn[d], 1, red8);
        omxr[d] = blk_red(omx[d], 2, red8);
    }
    float pszr = blk_red(psz, 0, red8);
    float recr = blk_red(rec, 0, red8);

    float zmin = omnr[2], zmax = omxr[2];
    float zr = zmax - zmin;
    float loThr = zmin + 0.15f * zr;
    float hiThr = zmax - 0.15f * zr;
    float en = 0.0f, ec = 0.0f;
    for (int i = tid; i < NPG; i += 256) {
        float oz = ob[3 * i + 2];
        if (oz <= loThr || oz >= hiThr) {
            float dx = pb[3 * i + 0] - ob[3 * i + 0];
            float dy = pb[3 * i + 1] - ob[3 * i + 1];
            float dz = pb[3 * i + 2] - oz;
            en += dx * dx + dy * dy + dz * dz;
            ec += 1.0f;
        }
    }
    en = blk_red(en, 0, red8);
    ec = blk_red(ec, 0, red8);

    if (tid == 0) {
        atomicAdd(&ws[0], recr);
        float vol = 0.0f;
        float rr[3];
        #pragma unroll
        for (int d = 0; d < 3; ++d) {
            float pd = pmxr[d] - pmnr[d];
            float od = omxr[d] - omnr[d];
            rr[d] = (pd - od) / (od + 1e-8f);
        }
        vol += fmaxf(rr[0] - 0.02f, 0.0f);
        vol += fmaxf(rr[1] - 0.02f, 0.0f);
        vol += 2.0f * fmaxf(rr[2], 0.0f);
        atomicAdd(&ws[1], vol);
        atomicAdd(&ws[2], en);
        atomicAdd(&ws[3], ec);
        float v = nf[(size_t)b * NPG * NF + (NF - 3)];
        float zcom  = pszr * (1.0f / NPG);
        float zpmin = pmnr[2];
        float zpr   = pmxr[2] - pmnr[2];
        float znorm = (zcom - zpmin) / (zpr + 1e-8f);
        float target = (v > 0.0f) ? 0.6f : 0.4f;
        float dzt = znorm - target;
        float contrib = (fabsf(v) >= 1e-6f) ? dzt * dzt : 0.0f;
        atomicAdd(&ws[5], contrib);
    }
}

// One 16x16 tile of the pairwise-violation sum via V_WMMA_F32_16X16X4_F32.
// A row i = (x,y,z,|p_i|^2), B col j = (-2x,-2y,-2z,1) => D = |p_i|^2 - 2 p_i.p_j,
// then d2 = D + |p_j|^2 (per-lane scalar). DIAG tiles mask out i==j.
template <bool DIAG>
__device__ __forceinline__ float tile_contrib(const float4* __restrict__ pts,
                                              int tm, int tn, int lmod, bool hiH) {
    float4 pa = pts[tm * 16 + lmod];
    float4 pc = pts[tn * 16 + lmod];
    v2f A, Bv;
    A.x = hiH ? pa.z : pa.x;            // lanes 0-15: K=0,1 ; lanes 16-31: K=2,3
    A.y = hiH ? pa.w : pa.y;
    Bv.x = hiH ? (-2.0f * pc.z) : (-2.0f * pc.x);
    Bv.y = hiH ? 1.0f            : (-2.0f * pc.y);
    v8f C = {};
    C = __builtin_amdgcn_wmma_f32_16x16x4_f32(
            false, A, false, Bv, (short)0, C, false, false);
    float sqj = pc.w;
    float acc = 0.0f;
    #pragma unroll
    for (int k = 0; k < 8; ++k) {
        float d2 = C[k] + sqj;
        float d  = __builtin_amdgcn_sqrtf(fmaxf(d2, 1e-12f));
        float tt = fmaxf(2.9f - d, 0.0f);
        float c2 = tt * tt;
        if (DIAG) {
            int M = k + (hiH ? 8 : 0);
            if (M == lmod) c2 = 0.0f;    // exclude i==j
        }
        acc += c2;
    }
    return acc;
}

__global__ void collapse_kernel(const float* __restrict__ pred,
                                float* __restrict__ ws) {
    __shared__ float4 pts[NPG];
    __shared__ float red8[8];
    const int b     = blockIdx.x / SLICES;
    const int slice = blockIdx.x % SLICES;
    const int tid   = threadIdx.x;
    const float* pb = pred + (size_t)b * NPG * 3;

    for (int i = tid; i < NPG; i += 256) {
        float x = pb[3 * i + 0];
        float y = pb[3 * i + 1];
        float z = pb[3 * i + 2];
        pts[i] = make_float4(x, y, z, x * x + y * y + z * z);
    }
    __syncthreads();

    const int lane = tid & 31;
    const int lmod = lane & 15;
    const bool hiH = lane >= 16;
    // wave-uniform slot id, forced scalar so loop induction lives in SGPRs
    const int myslot = __builtin_amdgcn_readfirstlane(slice * 8 + (tid >> 5)); // 0..127

    float acc_off = 0.0f, acc_diag = 0.0f;

    // ---- strict upper triangle: 2016 tiles, direct (tm,tn) from tile index ----
    // tiles before row m: S(m) = m*(127-m)/2 ;  tm = floor(63.5 - sqrt(63.5^2 - 2t))
    #pragma clang loop unroll_count(2)
    for (int t = myslot; t < 2016; t += 128) {
        float disc = 4032.25f - 2.0f * (float)t;
        int tm = (int)(63.5f - __builtin_amdgcn_sqrtf(disc));
        if ((tm + 1) * (127 - (tm + 1)) / 2 <= t) ++tm;   // sqrt rounded up
        if (tm * (127 - tm) / 2 > t)              --tm;   // sqrt rounded down
        int tn = tm + 1 + (t - tm * (127 - tm) / 2);
        acc_off += tile_contrib<false>(pts, tm, tn, lmod, hiH);
    }

    // ---- diagonal tiles: one per slot for slots 0..63 (scalar branch) ----
    if (myslot < 64)
        acc_diag += tile_contrib<true>(pts, myslot, myslot, lmod, hiH);

    float s = blk_red(2.0f * acc_off + acc_diag, 0, red8);
    if (tid == 0)
        atomicAdd(&ws[4], s * (1.0f / ((float)NPG * (float)(NPG - 1))));
}

__global__ void finalize_kernel(const float* __restrict__ ws, float* __restrict__ out) {
    if (threadIdx.x == 0 && blockIdx.x == 0) {
        float recon = ws[0] * (1.0f / ((float)NTOT * 3.0f));
        float vol   = ws[1] * (1.0f / (float)BATCHES);
        float elec  = ws[2] / (ws[3] * 3.0f);
        float coll  = ws[4] * (1.0f / (float)BATCHES);
        float field = ws[5] * (1.0f / (float)BATCHES);
        float total = 1.0f * recon + 10.0f * vol + 50.0f * elec
                    + 5.0f * coll + 2.0f * field;
        out[0] = total;
        out[1] = recon;
        out[2] = vol;
        out[3] = elec;
        out[4] = coll;
        out[5] = field;
    }
}

extern "C" void kernel_launch(void* const* d_in, const int* in_sizes, int n_in,
                              void* d_out, int out_size, void* d_ws, size_t ws_size,
                              hipStream_t stream) {
    const float* pred = (const float*)d_in[0];
    const float* tru  = (const float*)d_in[1];
    const float* orig = (const float*)d_in[2];
    const float* nf   = (const float*)d_in[3];
    // d_in[4] (batch_vector) unused: batches are contiguous by construction
    float* ws  = (float*)d_ws;
    float* out = (float*)d_out;

    init_kernel<<<dim3(1), dim3(32), 0, stream>>>(ws);
    stats_kernel<<<dim3(BATCHES), dim3(256), 0, stream>>>(pred, tru, orig, nf, ws);
    collapse_kernel<<<dim3(BATCHES * SLICES), dim3(256), 0, stream>>>(pred, ws);
    finalize_kernel<<<dim3(1), dim3(32), 0, stream>>>(ws, out);
}